// DummyModel_3719441679142
// MI455X (gfx1250) — compile-verified
//
#include <hip/hip_runtime.h>

// Problem constants (from the reference): seq [256,4096] int32, rand [256,4096,21] f32
#define NALPH      21
#define UNK        20
#define TPB        256                    // 8 wave32s per block
#define TILE       256                    // positions per block (1 per thread)
#define TILE_FLTS  (TILE * NALPH)         // 5376 floats
#define TILE_BYTES (TILE_FLTS * 4)        // 21504 bytes, multiple of 16
#define CHUNKS     (TILE_BYTES / 16)      // 1344 b128 chunks per tile

// The async-LDS builtins take pointers to 4xi32 vectors (per the round-1
// diagnostic: parameter type is 'int __attribute__((vector_size(16)))*').
typedef int v4i __attribute__((ext_vector_type(4)));
typedef __attribute__((address_space(1))) v4i gv4i_t;   // global
typedef __attribute__((address_space(3))) v4i lv4i_t;   // LDS

__global__ __launch_bounds__(TPB) void onehot_unk_kernel(
    const int*   __restrict__ seq,
    const float* __restrict__ rv,
    float*       __restrict__ out)
{
    // LDS staging tile: row stride 21 dwords is coprime with 64 banks ->
    // the per-thread row writes below are bank-conflict free.
    __shared__ __align__(16) float tile[TILE_FLTS];

    const int t   = threadIdx.x;
    const int pos = blockIdx.x * TILE + t;          // one position per thread

    // ---- 1) token (coalesced b32 load) -------------------------------------
    const int tok = seq[pos];

    // ---- 2) build this position's 21 outputs in LDS ------------------------
    float* row = &tile[t * NALPH];

    if (tok == UNK) {
        // Only ~4.8% of lanes take this path; EXEC-masked loads mean the
        // rand_vals traffic is ~1/21 of the tensor instead of all 88 MB.
        const float* src = rv + (size_t)pos * NALPH;
        float v[NALPH];
        float s = 0.0f;
#pragma unroll
        for (int c = 0; c < NALPH; ++c) { v[c] = src[c]; s += v[c]; }
        const float inv = 1.0f / s;                 // one precise divide
#pragma unroll
        for (int c = 0; c < NALPH; ++c) row[c] = v[c] * inv;
    } else {
#pragma unroll
        for (int c = 0; c < NALPH; ++c) row[c] = 0.0f;
        row[tok] = 1.0f;
    }

    __syncthreads();   // LDS tile complete (s_barrier + dscnt wait)

    // ---- 3) drain tile -> global with coalesced 16B async stores -----------
    // Tile base is 21504-byte aligned in the output, so every chunk is a
    // 16B-aligned b128. Each wave-level store covers 512 contiguous bytes.
    char* gbase = (char*)out + (size_t)blockIdx.x * TILE_BYTES;

#pragma unroll
    for (int c = t; c < CHUNKS; c += TPB) {
#if __has_builtin(__builtin_amdgcn_global_store_async_from_lds_b128)
        __builtin_amdgcn_global_store_async_from_lds_b128(
            (gv4i_t*)(gbase + (size_t)c * 16),
            (lv4i_t*)((char*)tile + (size_t)c * 16),
            /*imm offset=*/0, /*cpol=*/0);
#else
        // Fallback: VGPR round-trip, still perfectly coalesced b128s.
        ((float4*)gbase)[c] = ((const float4*)tile)[c];
#endif
    }

#if __has_builtin(__builtin_amdgcn_s_wait_asynccnt)
    __builtin_amdgcn_s_wait_asynccnt(0);
#else
    asm volatile("s_wait_asynccnt 0x0" ::: "memory");
#endif
}

extern "C" void kernel_launch(void* const* d_in, const int* in_sizes, int n_in,
                              void* d_out, int out_size, void* d_ws, size_t ws_size,
                              hipStream_t stream) {
    const int*   seq = (const int*)  d_in[0];   // [NB, L] int32
    const float* rv  = (const float*)d_in[1];   // [NB, L, 21] f32
    float*       out = (float*)      d_out;     // [NB, L, 21] f32

    const int npos    = in_sizes[0];            // NB*L = 1,048,576
    const int nblocks = npos / TILE;            // 4096

    onehot_unk_kernel<<<nblocks, TPB, 0, stream>>>(seq, rv, out);
}